// mat_GRU_cell_26938034880815
// MI455X (gfx1250) — compile-verified
//
#include <hip/hip_runtime.h>
#include <hip/hip_bf16.h>

// Problem constants (fixed by the reference)
#define BB 64
#define NN 4096
#define FF 512
#define KK 128

typedef float v2f __attribute__((ext_vector_type(2)));
typedef float v8f __attribute__((ext_vector_type(8)));

__device__ __forceinline__ float sigmoidf_(float x) { return 1.0f / (1.0f + expf(-x)); }

// ---------------- WMMA f32 16x16x4 helpers ----------------
// D = A(16x4) x B(4x16) + C(16x16), wave32, f32 everywhere.
__device__ __forceinline__ v8f wmma4(v2f a, v2f b, v8f c) {
  return __builtin_amdgcn_wmma_f32_16x16x4_f32(
      /*neg_a=*/false, a, /*neg_b=*/false, b,
      /*c_mod=*/(short)0, c, /*reuse_a=*/false, /*reuse_b=*/false);
}

// A-frag (and B-frag sourced from a transposed matrix): row-major [*,128].
// lane = {m = lane&15, hi = lane>>4}:  {P[row][k+2*hi], P[row][k+2*hi+1]}
__device__ __forceinline__ v2f frag_rowk(const float* __restrict__ p, int row, int k, int hi) {
  const float* q = p + row * KK + k + 2 * hi;
  v2f r; r.x = q[0]; r.y = q[1];
  return r;
}
// B-frag from a row-major [128,128] matrix used as-is (K along rows):
// {P[k+2*hi][col], P[k+2*hi+1][col]}
__device__ __forceinline__ v2f frag_colk(const float* __restrict__ p, int col, int k, int hi) {
  const float* q = p + (k + 2 * hi) * KK + col;
  v2f r; r.x = q[0]; r.y = q[KK];
  return r;
}
__device__ __forceinline__ v8f bias_frag(const float* __restrict__ bia, int ti, int tj, int m, int hi) {
  v8f c;
#pragma unroll
  for (int v = 0; v < 8; ++v) c[v] = bia[(ti * 16 + v + 8 * hi) * KK + tj * 16 + m];
  return c;
}

// ---------------- K1: scorer = tanh(ht @ Wmap^T + b), plus 1/||scorer|| ----------------
__global__ void scorer_kernel(const float* __restrict__ ht, const float* __restrict__ Wmap,
                              const float* __restrict__ bmap, float* __restrict__ scorer_out,
                              float* __restrict__ inv_norm) {
  __shared__ __align__(16) float hts[FF];
  __shared__ float red[16];
  const int b = blockIdx.x;
  const int f = threadIdx.x;  // 512 threads
  hts[f] = ht[b * FF + f];
  __syncthreads();
  const float4* w4 = (const float4*)(Wmap + (size_t)f * FF);
  const float4* h4 = (const float4*)hts;
  float acc = bmap[f];
#pragma unroll 4
  for (int d = 0; d < FF / 4; ++d) {
    float4 wv = w4[d], hv = h4[d];
    acc += wv.x * hv.x + wv.y * hv.y + wv.z * hv.z + wv.w * hv.w;
  }
  float s = tanhf(acc);
  scorer_out[b * FF + f] = s;
  float ss = s * s;
#pragma unroll
  for (int off = 16; off; off >>= 1) ss += __shfl_xor(ss, off, 32);
  if ((f & 31) == 0) red[f >> 5] = ss;
  __syncthreads();
  if (f == 0) {
    float tot = 0.f;
#pragma unroll
    for (int i = 0; i < 16; ++i) tot += red[i];
    inv_norm[b] = rsqrtf(tot);
  }
}

// ---------------- K2: scores[b,n] = dot(Z[b,n,:], scorer[b,:]) * inv_norm[b] ----------------
// One wave per (b,n) row; b128 coalesced streaming of the 512MB prev_Z tensor.
__global__ void scores_kernel(const float* __restrict__ Z, const float* __restrict__ scorer,
                              const float* __restrict__ inv_norm, float* __restrict__ scores) {
  const int gt = blockIdx.x * blockDim.x + threadIdx.x;
  const int wave = gt >> 5;
  const int lane = threadIdx.x & 31;
  const int b = wave >> 12;        // 4096 rows per batch
  const int n = wave & (NN - 1);
  const float4* z4 = (const float4*)(Z + ((size_t)b * NN + n) * FF);
  const float4* s4 = (const float4*)(scorer + (size_t)b * FF);
  float acc = 0.f;
#pragma unroll
  for (int j = 0; j < 4; ++j) {
    float4 zv = z4[lane + 32 * j];
    float4 sv = s4[lane + 32 * j];
    acc += zv.x * sv.x + zv.y * sv.y + zv.z * sv.z + zv.w * sv.w;
  }
#pragma unroll
  for (int off = 16; off; off >>= 1) acc += __shfl_xor(acc, off, 32);
  if (lane == 0) scores[(size_t)b * NN + n] = acc * inv_norm[b];
}

// ---------------- K3: log-softmax stats per batch: logZ, entropy ----------------
__global__ void softmax_stats_kernel(const float* __restrict__ scores,
                                     float* __restrict__ logZ, float* __restrict__ entropy_out) {
  __shared__ float redA[16];
  __shared__ float redB[16];
  __shared__ float bcast;
  const int b = blockIdx.x;
  const int t = threadIdx.x;  // 512 threads, 8 elems each
  const float* s = scores + (size_t)b * NN;
  float v[8];
  float lmax = -3.4e38f;
#pragma unroll
  for (int i = 0; i < 8; ++i) { v[i] = s[t + 512 * i]; lmax = fmaxf(lmax, v[i]); }
#pragma unroll
  for (int off = 16; off; off >>= 1) lmax = fmaxf(lmax, __shfl_xor(lmax, off, 32));
  if ((t & 31) == 0) redA[t >> 5] = lmax;
  __syncthreads();
  if (t == 0) {
    float x = redA[0];
#pragma unroll
    for (int i = 1; i < 16; ++i) x = fmaxf(x, redA[i]);
    bcast = x;
  }
  __syncthreads();
  const float mx = bcast;
  float se = 0.f, ssum = 0.f;
#pragma unroll
  for (int i = 0; i < 8; ++i) {
    float e = expf(v[i] - mx);
    se += e;
    ssum += e * v[i];
  }
#pragma unroll
  for (int off = 16; off; off >>= 1) {
    se += __shfl_xor(se, off, 32);
    ssum += __shfl_xor(ssum, off, 32);
  }
  if ((t & 31) == 0) { redA[t >> 5] = se; redB[t >> 5] = ssum; }
  __syncthreads();
  if (t == 0) {
    float te = 0.f, ts = 0.f;
#pragma unroll
    for (int i = 0; i < 16; ++i) { te += redA[i]; ts += redB[i]; }
    float lz = mx + logf(te);
    logZ[b] = lz;
    entropy_out[b] = lz - ts / te;  // H = logZ - E[s]
  }
}

// ---------------- K4: exact sorted top-K by rank counting + policy score ----------------
__global__ void topk_kernel(const float* __restrict__ scores, const float* __restrict__ logZ,
                            float* __restrict__ vals_out, int* __restrict__ idx_int_out,
                            float* __restrict__ idx_f_out, float* __restrict__ policy_out) {
  __shared__ float s[NN];     // 16 KB
  __shared__ float tv[KK];
  __shared__ float red[16];
  const int b = blockIdx.x;
  const int t = threadIdx.x;  // 512 threads
#pragma unroll
  for (int i = 0; i < 8; ++i) s[t + 512 * i] = scores[(size_t)b * NN + t + 512 * i];
  __syncthreads();
#pragma unroll 2
  for (int i = 0; i < 8; ++i) {
    const int n = t + 512 * i;
    const float sn = s[n];
    int rank = 0;
    for (int m = 0; m < NN; ++m) {
      const float sm = s[m];
      rank += (sm > sn) || (sm == sn && m < n);  // value desc, index asc (matches top_k)
    }
    if (rank < KK) {
      tv[rank] = sn;
      vals_out[b * KK + rank] = sn;
      idx_int_out[b * KK + rank] = n;
      idx_f_out[b * KK + rank] = (float)n;
    }
  }
  __syncthreads();
  float x = (t < KK) ? tv[t] : 0.f;
#pragma unroll
  for (int off = 16; off; off >>= 1) x += __shfl_xor(x, off, 32);
  if ((t & 31) == 0) red[t >> 5] = x;
  __syncthreads();
  if (t == 0) {
    float tot = 0.f;
#pragma unroll
    for (int i = 0; i < 16; ++i) tot += red[i];
    policy_out[b] = tot * (1.0f / KK) - logZ[b];
  }
}

// ---------------- K5: gather  X[b,k,j] = Z[b, idx_k, j] * tanh(val_k),  j<128 ----------------
__global__ void gather_kernel(const float* __restrict__ Z, const int* __restrict__ idx,
                              const float* __restrict__ vals, float* __restrict__ X) {
  const int b = blockIdx.x;
  const int t = threadIdx.x;      // 512
  const int j = t & (KK - 1);
  const int ko = t >> 7;          // 0..3
  for (int k = ko; k < KK; k += 4) {
    const int n = idx[b * KK + k];
    const float g = tanhf(vals[b * KK + k]);
    X[((size_t)b * KK + k) * KK + j] = Z[((size_t)b * NN + n) * FF + j] * g;
  }
}

// ---------------- K6a: update & reset gates via WMMA; emit update and R = reset.*Q ----------
__global__ void gru_gates_kernel(const float* __restrict__ Wu, const float* __restrict__ Uu,
                                 const float* __restrict__ bu,
                                 const float* __restrict__ Wr, const float* __restrict__ Ur,
                                 const float* __restrict__ br,
                                 const float* __restrict__ X, const float* __restrict__ Q,
                                 float* __restrict__ upd, float* __restrict__ RQ) {
  const int lane = threadIdx.x & 31;
  const int wv = threadIdx.x >> 5;           // 8 waves / block
  const int m = lane & 15, hi = lane >> 4;
  const int b = blockIdx.x >> 3;
  const int tile = ((blockIdx.x & 7) << 3) | wv;  // 0..63
  const int ti = tile >> 3, tj = tile & 7;
  const float* Xb = X + (size_t)b * (KK * KK);
  const float* Qb = Q + (size_t)b * (KK * KK);
  const int arow = ti * 16 + m;
  const int bcol = tj * 16 + m;

  v8f cu = bias_frag(bu, ti, tj, m, hi);
  v8f cr = bias_frag(br, ti, tj, m, hi);
  // W @ z_topk : B[j][n] = X[n][j]  -> row-contiguous per lane
#pragma unroll 8
  for (int k = 0; k < KK; k += 4) {
    v2f bX = frag_rowk(Xb, bcol, k, hi);
    cu = wmma4(frag_rowk(Wu, arow, k, hi), bX, cu);
    cr = wmma4(frag_rowk(Wr, arow, k, hi), bX, cr);
  }
  // U @ prev_Q : B = Q as-is
#pragma unroll 8
  for (int k = 0; k < KK; k += 4) {
    v2f bQ = frag_colk(Qb, bcol, k, hi);
    cu = wmma4(frag_rowk(Uu, arow, k, hi), bQ, cu);
    cr = wmma4(frag_rowk(Ur, arow, k, hi), bQ, cr);
  }
  const size_t base = (size_t)b * (KK * KK) + tj * 16 + m;
#pragma unroll
  for (int v = 0; v < 8; ++v) {
    const int row = ti * 16 + v + 8 * hi;
    const float u = sigmoidf_(cu[v]);
    const float r = sigmoidf_(cr[v]);
    upd[base + (size_t)row * KK] = u;
    RQ[base + (size_t)row * KK] = r * Qb[row * KK + tj * 16 + m];
  }
}

// ---------------- K6b: h_cap via WMMA, blend to new_Q ----------------
__global__ void gru_hnew_kernel(const float* __restrict__ Wh, const float* __restrict__ Uh,
                                const float* __restrict__ bh,
                                const float* __restrict__ X, const float* __restrict__ RQ,
                                const float* __restrict__ Q, const float* __restrict__ upd,
                                float* __restrict__ newQ) {
  const int lane = threadIdx.x & 31;
  const int wv = threadIdx.x >> 5;
  const int m = lane & 15, hi = lane >> 4;
  const int b = blockIdx.x >> 3;
  const int tile = ((blockIdx.x & 7) << 3) | wv;
  const int ti = tile >> 3, tj = tile & 7;
  const float* Xb = X + (size_t)b * (KK * KK);
  const float* Rb = RQ + (size_t)b * (KK * KK);
  const float* Qb = Q + (size_t)b * (KK * KK);
  const int arow = ti * 16 + m;
  const int bcol = tj * 16 + m;

  v8f ch = bias_frag(bh, ti, tj, m, hi);
#pragma unroll 8
  for (int k = 0; k < KK; k += 4)
    ch = wmma4(frag_rowk(Wh, arow, k, hi), frag_rowk(Xb, bcol, k, hi), ch);
#pragma unroll 8
  for (int k = 0; k < KK; k += 4)
    ch = wmma4(frag_rowk(Uh, arow, k, hi), frag_colk(Rb, bcol, k, hi), ch);

  const size_t base = (size_t)b * (KK * KK) + tj * 16 + m;
#pragma unroll
  for (int v = 0; v < 8; ++v) {
    const int row = ti * 16 + v + 8 * hi;
    const float h = tanhf(ch[v]);
    const float u = upd[base + (size_t)row * KK];
    const float q = Qb[row * KK + tj * 16 + m];
    newQ[base + (size_t)row * KK] = (1.f - u) * q + u * h;
  }
}

extern "C" void kernel_launch(void* const* d_in, const int* in_sizes, int n_in,
                              void* d_out, int out_size, void* d_ws, size_t ws_size,
                              hipStream_t stream) {
  const float* prev_Q = (const float*)d_in[0];
  const float* prev_Z = (const float*)d_in[1];
  const float* ht     = (const float*)d_in[2];
  const float* W_map  = (const float*)d_in[3];
  const float* b_map  = (const float*)d_in[4];
  const float* W_u = (const float*)d_in[5];
  const float* U_u = (const float*)d_in[6];
  const float* b_u = (const float*)d_in[7];
  const float* W_r = (const float*)d_in[8];
  const float* U_r = (const float*)d_in[9];
  const float* b_r = (const float*)d_in[10];
  const float* W_h = (const float*)d_in[11];
  const float* U_h = (const float*)d_in[12];
  const float* b_h = (const float*)d_in[13];

  // outputs, concatenated flat: new_Q, policy_score, scorer, entropy, topk_idx
  float* out = (float*)d_out;
  float* o_newQ    = out;                                 // 64*128*128
  float* o_policy  = o_newQ + (size_t)BB * KK * KK;       // 64
  float* o_scorer  = o_policy + BB;                       // 64*512
  float* o_entropy = o_scorer + (size_t)BB * FF;          // 64
  float* o_topk    = o_entropy + BB;                      // 64*128 (as float)

  // workspace layout
  float* ws        = (float*)d_ws;
  float* w_scores  = ws;                                  // B*N
  float* w_invn    = w_scores + (size_t)BB * NN;          // B
  float* w_logZ    = w_invn + BB;                         // B
  float* w_vals    = w_logZ + BB;                         // B*K
  int*   w_idx     = (int*)(w_vals + (size_t)BB * KK);    // B*K
  float* w_X       = (float*)(w_idx + (size_t)BB * KK);   // B*K*K
  float* w_upd     = w_X + (size_t)BB * KK * KK;          // B*K*K
  float* w_RQ      = w_upd + (size_t)BB * KK * KK;        // B*K*K

  scorer_kernel<<<BB, FF, 0, stream>>>(ht, W_map, b_map, o_scorer, w_invn);
  scores_kernel<<<(BB * NN) / 8, 256, 0, stream>>>(prev_Z, o_scorer, w_invn, w_scores);
  softmax_stats_kernel<<<BB, 512, 0, stream>>>(w_scores, w_logZ, o_entropy);
  topk_kernel<<<BB, 512, 0, stream>>>(w_scores, w_logZ, w_vals, w_idx, o_topk, o_policy);
  gather_kernel<<<BB, 512, 0, stream>>>(prev_Z, w_idx, w_vals, w_X);
  gru_gates_kernel<<<BB * 8, 256, 0, stream>>>(W_u, U_u, b_u, W_r, U_r, b_r,
                                               w_X, prev_Q, w_upd, w_RQ);
  gru_hnew_kernel<<<BB * 8, 256, 0, stream>>>(W_h, U_h, b_h, w_X, w_RQ,
                                              prev_Q, w_upd, o_newQ);
}